// BackgroundNoiseLayer_27831388078548
// MI455X (gfx1250) — compile-verified
//
#include <hip/hip_runtime.h>
#include <hip/hip_bf16.h>

// Problem constants (from reference)
#define N_NEURONS 65536
#define N_BKG 100
#define N_SYN 5
#define NNZ 655360
#define SEQ_LEN 500
#define SEQ_PAD 512                  // pad T to 512 = 32 tiles of 16
#define NCOL (N_NEURONS * N_SYN)     // 327680 output columns
#define SPIKE_P 0.25f

typedef __attribute__((ext_vector_type(2))) float v2f;
typedef __attribute__((ext_vector_type(8))) float v8f;

// ---------------------------------------------------------------------------
// Kernel 1: zero the densified weight matrix Wt[u][ncol] (131 MB) in d_ws
// ---------------------------------------------------------------------------
__global__ __launch_bounds__(256) void zero_wt_kernel(float4* __restrict__ p, int n4) {
    int i = blockIdx.x * blockDim.x + threadIdx.x;
    if (i < n4) p[i] = make_float4(0.f, 0.f, 0.f, 0.f);
}

// ---------------------------------------------------------------------------
// Kernel 2: spikes_pad[u][t] = (t < 500 && noise_u[t*100+u] < 0.25) ? 1 : 0
// Layout [U][T_pad] so A-fragment loads (lane = t) are contiguous 64B runs.
// ---------------------------------------------------------------------------
__global__ __launch_bounds__(256) void spikes_kernel(const float* __restrict__ noise,
                                                     float* __restrict__ sp) {
    int i = blockIdx.x * blockDim.x + threadIdx.x;   // i = u*512 + t
    if (i >= N_BKG * SEQ_PAD) return;
    int u = i >> 9;
    int t = i & (SEQ_PAD - 1);
    float v = 0.f;
    if (t < SEQ_LEN) v = (noise[t * N_BKG + u] < SPIKE_P) ? 1.f : 0.f;
    sp[i] = v;
}

// ---------------------------------------------------------------------------
// Kernel 3: scatter-add edges into Wt[u][row*5 + r]
// ---------------------------------------------------------------------------
__global__ __launch_bounds__(256) void scatter_kernel(const float* __restrict__ bw,
                                                      const float* __restrict__ factors,
                                                      const int* __restrict__ row_idx,
                                                      const int* __restrict__ col_idx,
                                                      float* __restrict__ Wt) {
    int e = blockIdx.x * blockDim.x + threadIdx.x;
    if (e >= NNZ) return;
    float w = bw[e];
    int r = row_idx[e];
    int c = col_idx[e];
    size_t base = (size_t)c * NCOL + (size_t)r * N_SYN;
    #pragma unroll
    for (int i = 0; i < N_SYN; ++i)
        atomicAdd(&Wt[base + i], w * factors[(size_t)e * N_SYN + i]);
}

// ---------------------------------------------------------------------------
// Kernel 4: out[t][col] = sum_u spikes[t][u] * Wt[u][col]
// One wave32 per 16-column tile. V_WMMA_F32_16X16X4_F32, K = 100 = 25 x 4.
// B slab (100x16 f32) held in 50 VGPRs for the whole column tile; sweep all
// 32 row tiles of the (padded) 512x100 spike matrix.
// ---------------------------------------------------------------------------
__global__ __launch_bounds__(256) void wmma_gemm_kernel(const float* __restrict__ sp,  // [100][512]
                                                        const float* __restrict__ Wt,  // [100][NCOL]
                                                        float* __restrict__ out) {     // [500][NCOL]
    const int lane = threadIdx.x & 31;
    const int g    = lane >> 4;      // lane group: selects K pair within a 4-step
    const int ln   = lane & 15;      // M (for A) / N (for B,C,D) within tile
    const int wave = blockIdx.x * (blockDim.x >> 5) + (threadIdx.x >> 5);
    const size_t n_base = (size_t)wave * 16;

    // Load B fragments: b[k].x = Wt[4k+2g][n], b[k].y = Wt[4k+2g+1][n], n = n_base+ln
    v2f b[25];
    #pragma unroll
    for (int k = 0; k < 25; ++k) {
        int u0 = 4 * k + 2 * g;
        b[k].x = Wt[(size_t)u0       * NCOL + n_base + ln];
        b[k].y = Wt[(size_t)(u0 + 1) * NCOL + n_base + ln];
    }

    for (int m = 0; m < SEQ_PAD / 16; ++m) {
        const int t = m * 16 + ln;   // A row for this lane (padded rows read 0)
        v8f acc = {};
        #pragma unroll
        for (int k = 0; k < 25; ++k) {
            int u0 = 4 * k + 2 * g;
            v2f a;
            a.x = sp[u0       * SEQ_PAD + t];
            a.y = sp[(u0 + 1) * SEQ_PAD + t];
            // (neg_a, A, neg_b, B, c_mod, C, reuse_a, reuse_b)
            acc = __builtin_amdgcn_wmma_f32_16x16x4_f32(
                false, a, false, b[k], (short)0, acc, false, false);
        }
        // C/D layout: VGPR j, lane -> row = 16m + j + 8g, col = n_base + ln
        #pragma unroll
        for (int j = 0; j < 8; ++j) {
            int row = m * 16 + j + 8 * g;
            if (row < SEQ_LEN)
                out[(size_t)row * NCOL + n_base + ln] = acc[j];
        }
    }
}

// ---------------------------------------------------------------------------
// Host launcher
// ---------------------------------------------------------------------------
extern "C" void kernel_launch(void* const* d_in, const int* in_sizes, int n_in,
                              void* d_out, int out_size, void* d_ws, size_t ws_size,
                              hipStream_t stream) {
    const float* noise_u  = (const float*)d_in[0];   // 500*100
    const float* bkg_w    = (const float*)d_in[1];   // NNZ
    const float* factors  = (const float*)d_in[2];   // NNZ*5
    const int*   row_idx  = (const int*)d_in[3];     // NNZ
    const int*   col_idx  = (const int*)d_in[4];     // NNZ
    float*       out      = (float*)d_out;           // 500 * 327680

    // Workspace layout: spikes_pad (100*512 f32 = 204800 B, 16B aligned),
    // then Wt (100*327680 f32 = 131.072 MB). Total ~131.3 MB.
    float* spikes = (float*)d_ws;
    float* Wt     = spikes + (size_t)N_BKG * SEQ_PAD;

    // 1) zero Wt
    {
        int n4 = (N_BKG * NCOL) / 4;                 // 8,192,000 float4
        int blocks = (n4 + 255) / 256;
        zero_wt_kernel<<<blocks, 256, 0, stream>>>((float4*)Wt, n4);
    }
    // 2) spike matrix (padded, transposed)
    {
        int n = N_BKG * SEQ_PAD;
        spikes_kernel<<<(n + 255) / 256, 256, 0, stream>>>(noise_u, spikes);
    }
    // 3) scatter edges into Wt
    {
        scatter_kernel<<<(NNZ + 255) / 256, 256, 0, stream>>>(bkg_w, factors, row_idx,
                                                              col_idx, Wt);
    }
    // 4) WMMA GEMM: 20480 column tiles, 8 waves/block -> 2560 blocks
    {
        int n_tiles = NCOL / 16;                     // 20480
        int blocks  = n_tiles / 8;                   // 2560
        wmma_gemm_kernel<<<blocks, 256, 0, stream>>>(spikes, Wt, out);
    }
}